// QwenCausalSelfAttention_31885837205770
// MI455X (gfx1250) — compile-verified
//
#include <hip/hip_runtime.h>
#include <hip/hip_bf16.h>
#include <math.h>
#include <stdint.h>

typedef float    v2f __attribute__((ext_vector_type(2)));
typedef float    v8f __attribute__((ext_vector_type(8)));
typedef uint32_t v4u __attribute__((ext_vector_type(4)));
typedef uint32_t v8u __attribute__((ext_vector_type(8)));

#define B_   2
#define T_   2048
#define C_   896
#define H_   14
#define HKV_ 2
#define D_   64
#define NQKV (H_*D_ + 2*HKV_*D_)   /* 1152 */
#define M_   (B_*T_)               /* 4096 */

// f32 WMMA: D(16x16) = A(16x4) * B(4x16) + C, all f32.
// A layout: lanes 0-15 row M=lane (K=k+0,k+1 in two VGPRs); lanes 16-31 row M=lane-16 (K=k+2,k+3).
// B layout: lanes 0-15 col N=lane, K rows k+0/k+1; lanes 16-31 col N=lane-16, K rows k+2/k+3.
// C/D: VGPR r, lanes 0-15 -> (M=r, N=lane); lanes 16-31 -> (M=r+8, N=lane-16).
__device__ __forceinline__ v8f wmma_f32(v2f a, v2f b, v8f c) {
    return __builtin_amdgcn_wmma_f32_16x16x4_f32(false, a, false, b, (short)0, c, false, false);
}

// ---- Tensor Data Mover: async DMA of a 16x64 f32 tile (row stride 64 floats)
// from global memory into LDS. 2D tensor -> D# groups 0 and 1 only.
__device__ __forceinline__ void tdm_load_tile16x64(const float* gsrc, uint32_t lds_byte_addr) {
    const uint64_t ga = (uint64_t)(uintptr_t)gsrc;
    v4u g0;
    g0[0] = 1u;                               // count=1 (valid descriptor), no gather
    g0[1] = lds_byte_addr;                    // LDS byte address of tile start
    g0[2] = (uint32_t)ga;                     // global_addr[31:0]
    g0[3] = (uint32_t)(ga >> 32) | (2u << 30);// global_addr[56:32] | type=2 ("image")
    v8u g1;
    g1[0] = 2u << 16;                         // workgroup_mask=0, data_size=2 (4 bytes)
    g1[1] = (uint32_t)(64u) << 16;            // tensor_dim0[15:0]=64 -> bits[31:16]
    g1[2] = (uint32_t)(T_) << 16;             // tensor_dim0[31:16]=0 | tensor_dim1[15:0]=2048
    g1[3] = (uint32_t)(64u) << 16;            // tensor_dim1[31:16]=0 | tile_dim0=64
    g1[4] = 16u;                              // tile_dim1=16, tile_dim2=0
    g1[5] = 64u;                              // tensor_dim0_stride[31:0]=64
    g1[6] = 0u;                               // dim0_stride[47:32]=0, dim1_stride[15:0]=0
    g1[7] = 0u;                               // dim1_stride[47:16]=0
    asm volatile("tensor_load_to_lds %0, %1" :: "s"(g0), "s"(g1) : "memory");
}

// ---------------- K1: qkv = x @ [wq | wk | wv] ----------------
__global__ void qkv_gemm_kernel(const float* __restrict__ x,
                                const float* __restrict__ wq,
                                const float* __restrict__ wk,
                                const float* __restrict__ wv,
                                float* __restrict__ qkv)
{
    const int lane = threadIdx.x;
    const int nl   = lane & 15;
    const int lh   = lane >> 4;
    const int m0   = blockIdx.x * 16;
    const int n0   = (blockIdx.y * 4 + threadIdx.y) * 16;

    const float* wsrc; int ldw, ncol;
    if (n0 < H_*D_)              { wsrc = wq; ldw = H_*D_;   ncol = n0 + nl; }
    else if (n0 < H_*D_+HKV_*D_) { wsrc = wk; ldw = HKV_*D_; ncol = n0 - H_*D_ + nl; }
    else                         { wsrc = wv; ldw = HKV_*D_; ncol = n0 - (H_*D_+HKV_*D_) + nl; }

    v8f c = {};
    const float* arow = x + (size_t)(m0 + nl) * C_;
    for (int k = 0; k < C_; k += 4) {
        const int kr = k + 2*lh;
        v2f a = *(const v2f*)(arow + kr);
        v2f b;
        b.x = wsrc[(size_t)kr * ldw + ncol];
        b.y = wsrc[(size_t)(kr + 1) * ldw + ncol];
        c = wmma_f32(a, b, c);
    }
    float* crow = qkv + (size_t)(m0 + 8*lh) * NQKV + n0 + nl;
#pragma unroll
    for (int r = 0; r < 8; ++r)
        crow[(size_t)r * NQKV] = c[r];
}

// ---------------- K2: bias + RoPE + scatter into [B,H,T,D] ----------------
__global__ void rope_scatter_kernel(const float* __restrict__ qkv,
                                    const float* __restrict__ bq,
                                    const float* __restrict__ bk,
                                    const float* __restrict__ bv,
                                    const float* __restrict__ cosT,
                                    const float* __restrict__ sinT,
                                    float* __restrict__ qr,
                                    float* __restrict__ kr,
                                    float* __restrict__ vr)
{
    const int idx = blockIdx.x * blockDim.x + threadIdx.x;
    if (idx >= M_ * NQKV) return;
    const int m = idx / NQKV;
    const int n = idx - m * NQKV;
    const int b = m / T_;
    const int t = m - b * T_;

    auto fetch = [&](int nn) -> float {
        float bias = (nn < H_*D_) ? bq[nn]
                   : (nn < H_*D_+HKV_*D_) ? bk[nn - H_*D_]
                   : bv[nn - (H_*D_+HKV_*D_)];
        return qkv[(size_t)m * NQKV + nn] + bias;
    };

    const float val = fetch(n);
    const int d = n & 63;   // head boundaries are multiples of 64
    if (n < H_*D_ + HKV_*D_) {  // q or k: apply RoPE
        const float rot = (d < 32) ? -fetch(n + 32) : fetch(n - 32);
        const float cv = cosT[t * D_ + d];
        const float sv = sinT[t * D_ + d];
        const float out = val * cv + rot * sv;
        if (n < H_*D_) {
            const int h = n >> 6;
            qr[(((size_t)b*H_ + h)*T_ + t)*D_ + d] = out * 0.125f; // fold 1/sqrt(64)
        } else {
            const int hk = (n - H_*D_) >> 6;
            kr[(((size_t)b*HKV_ + hk)*T_ + t)*D_ + d] = out;
        }
    } else {
        const int hk = (n - (H_*D_+HKV_*D_)) >> 6;
        vr[(((size_t)b*HKV_ + hk)*T_ + t)*D_ + d] = val;
    }
}

// ---------------- K3: flash causal attention ----------------
// One wave per (b, h, 16-row q tile). K/V tiles stream through LDS via the
// Tensor Data Mover, double-buffered; s_wait_tensorcnt synchronizes.
__global__ void attn_kernel(const float* __restrict__ qr,
                            const float* __restrict__ kr,
                            const float* __restrict__ vr,
                            float* __restrict__ attn)
{
    __shared__ float Kl[2][16 * D_];
    __shared__ float Vl[2][16 * D_];
    __shared__ float pbuf[16 * 17];

    const int lane = threadIdx.x;
    const int nl = lane & 15;
    const int lh = lane >> 4;
    const int bh = blockIdx.x;       // b*H + h
    const int b  = bh / H_;
    const int h  = bh - b * H_;
    const int hk = h / (H_ / HKV_);
    const int i0 = blockIdx.y * 16;

    const float* qp = qr + (size_t)bh * T_ * D_;
    const float* kp = kr + ((size_t)b*HKV_ + hk) * T_ * D_;
    const float* vp = vr + ((size_t)b*HKV_ + hk) * T_ * D_;

    // LDS byte addresses for the TDM descriptors (flat addr low 32 bits = LDS offset)
    const uint32_t klbase = (uint32_t)(uintptr_t)(void*)&Kl[0][0];
    const uint32_t vlbase = (uint32_t)(uintptr_t)(void*)&Vl[0][0];
    const uint32_t tile_bytes = 16u * D_ * 4u;

    // Preload 16x64 Q tile into A-layout registers
    v2f aq[16];
#pragma unroll
    for (int kk = 0; kk < 16; ++kk)
        aq[kk] = *(const v2f*)(qp + (size_t)(i0 + nl) * D_ + kk*4 + 2*lh);

    v8f o0 = {}, o1 = {}, o2 = {}, o3 = {};
    float Mx[8], L[8];
#pragma unroll
    for (int r = 0; r < 8; ++r) { Mx[r] = -INFINITY; L[r] = 0.0f; }

    const int jt_end = i0 / 16;   // inclusive; diagonal tile

    // Prefetch first K/V tiles
    tdm_load_tile16x64(kp, klbase);
    tdm_load_tile16x64(vp, vlbase);

    for (int jt = 0; jt <= jt_end; ++jt) {
        const int j0 = jt * 16;
        const int cur = jt & 1;

        if (jt < jt_end) {
            // Prefetch next tiles into the other buffer, then wait for current
            // (TENSORcnt is in-order: <=2 outstanding => current tiles landed).
            tdm_load_tile16x64(kp + (size_t)(j0 + 16) * D_, klbase + (uint32_t)(1 - cur) * tile_bytes);
            tdm_load_tile16x64(vp + (size_t)(j0 + 16) * D_, vlbase + (uint32_t)(1 - cur) * tile_bytes);
            asm volatile("s_wait_tensorcnt 0x2" ::: "memory");
        } else {
            asm volatile("s_wait_tensorcnt 0x0" ::: "memory");
        }

        const float* kt = &Kl[cur][0];
        const float* vt = &Vl[cur][0];

        // S = Q @ K^T  (16 f32 WMMAs over D=64), K tile from LDS
        v8f s = {};
#pragma unroll
        for (int kk = 0; kk < 16; ++kk) {
            v2f kb = *(const v2f*)(kt + nl * D_ + kk*4 + 2*lh);
            s = wmma_f32(aq[kk], kb, s);
        }
        // Causal mask (diagonal tile only)
        if (jt == jt_end) {
            const int col = j0 + nl;
#pragma unroll
            for (int r = 0; r < 8; ++r) {
                const int row = i0 + r + 8*lh;
                if (col > row) s[r] = -1.0e9f;
            }
        }
        // Online softmax: rows live in (VGPR r, 16-lane half); reduce across 16 lanes
        float alpha[8];
#pragma unroll
        for (int r = 0; r < 8; ++r) {
            float v = s[r];
#pragma unroll
            for (int off = 8; off >= 1; off >>= 1)
                v = fmaxf(v, __shfl_xor(v, off, 32));
            const float mn = fmaxf(Mx[r], v);
            alpha[r] = __expf(Mx[r] - mn);
            Mx[r] = mn;
            const float p = __expf(s[r] - mn);
            float ps = p;
#pragma unroll
            for (int off = 8; off >= 1; off >>= 1)
                ps += __shfl_xor(ps, off, 32);
            L[r] = L[r] * alpha[r] + ps;
            pbuf[(r + 8*lh) * 17 + nl] = p;   // C-layout -> LDS for A-layout reload
        }
#pragma unroll
        for (int r = 0; r < 8; ++r) {
            o0[r] *= alpha[r]; o1[r] *= alpha[r]; o2[r] *= alpha[r]; o3[r] *= alpha[r];
        }
        // O += P @ V   (K dim = 16 keys -> 4 steps; 4 N-subtiles of 16), V tile from LDS
#pragma unroll
        for (int k0 = 0; k0 < 16; k0 += 4) {
            v2f pa;
            pa.x = pbuf[nl * 17 + k0 + 2*lh];
            pa.y = pbuf[nl * 17 + k0 + 2*lh + 1];
            const float* v0 = vt + (size_t)(k0 + 2*lh) * D_ + nl;
            v2f vb;
            vb.x = v0[0];   vb.y = v0[D_];        o0 = wmma_f32(pa, vb, o0);
            vb.x = v0[16];  vb.y = v0[D_ + 16];   o1 = wmma_f32(pa, vb, o1);
            vb.x = v0[32];  vb.y = v0[D_ + 32];   o2 = wmma_f32(pa, vb, o2);
            vb.x = v0[48];  vb.y = v0[D_ + 48];   o3 = wmma_f32(pa, vb, o3);
        }
    }
    // Normalize and store to [B, T, H*D]
    float* orow = attn + ((size_t)b * T_ + i0 + 8*lh) * (H_*D_) + h * D_ + nl;
#pragma unroll
    for (int r = 0; r < 8; ++r) {
        const float inv = 1.0f / L[r];
        orow[(size_t)r * (H_*D_) + 0]  = o0[r] * inv;
        orow[(size_t)r * (H_*D_) + 16] = o1[r] * inv;
        orow[(size_t)r * (H_*D_) + 32] = o2[r] * inv;
        orow[(size_t)r * (H_*D_) + 48] = o3[r] * inv;
    }
}

// ---------------- K4: out = attn @ wo ----------------
__global__ void out_gemm_kernel(const float* __restrict__ attn,
                                const float* __restrict__ wo,
                                float* __restrict__ out)
{
    const int lane = threadIdx.x;
    const int nl = lane & 15;
    const int lh = lane >> 4;
    const int m0 = blockIdx.x * 16;
    const int n0 = (blockIdx.y * 4 + threadIdx.y) * 16;

    v8f c = {};
    const float* arow = attn + (size_t)(m0 + nl) * (H_*D_);
    for (int k = 0; k < H_*D_; k += 4) {
        const int kr = k + 2*lh;
        v2f a = *(const v2f*)(arow + kr);
        v2f b;
        b.x = wo[(size_t)kr * C_ + n0 + nl];
        b.y = wo[(size_t)(kr + 1) * C_ + n0 + nl];
        c = wmma_f32(a, b, c);
    }
    float* crow = out + (size_t)(m0 + 8*lh) * C_ + n0 + nl;
#pragma unroll
    for (int r = 0; r < 8; ++r)
        crow[(size_t)r * C_] = c[r];
}

extern "C" void kernel_launch(void* const* d_in, const int* in_sizes, int n_in,
                              void* d_out, int out_size, void* d_ws, size_t ws_size,
                              hipStream_t stream) {
    (void)in_sizes; (void)n_in; (void)out_size; (void)ws_size;
    const float* x    = (const float*)d_in[0];
    const float* wq   = (const float*)d_in[1];
    const float* bq   = (const float*)d_in[2];
    const float* wk   = (const float*)d_in[3];
    const float* bk   = (const float*)d_in[4];
    const float* wv   = (const float*)d_in[5];
    const float* bv   = (const float*)d_in[6];
    const float* wo   = (const float*)d_in[7];
    const float* cosT = (const float*)d_in[8];
    const float* sinT = (const float*)d_in[9];
    // d_in[10] = mask (causality implemented directly)

    float* ws  = (float*)d_ws;
    float* qkv = ws;                                   // M_*NQKV floats (reused as attn buffer)
    float* qr  = qkv + (size_t)M_ * NQKV;              // B*H*T*D
    float* kr  = qr  + (size_t)B_ * H_ * T_ * D_;      // B*HKV*T*D
    float* vr  = kr  + (size_t)B_ * HKV_ * T_ * D_;    // B*HKV*T*D
    float* attn = qkv;                                 // safe reuse after rope_scatter

    qkv_gemm_kernel<<<dim3(M_/16, NQKV/16/4), dim3(32, 4), 0, stream>>>(x, wq, wk, wv, qkv);

    const int tot = M_ * NQKV;
    rope_scatter_kernel<<<(tot + 255)/256, 256, 0, stream>>>(qkv, bq, bk, bv, cosT, sinT, qr, kr, vr);

    attn_kernel<<<dim3(B_*H_, T_/16), dim3(32, 1), 0, stream>>>(qr, kr, vr, attn);

    out_gemm_kernel<<<dim3(M_/16, C_/16/4), dim3(32, 4), 0, stream>>>(attn, wo, (float*)d_out);
}